// ModelNew_21053929685603
// MI455X (gfx1250) — compile-verified
//
#include <hip/hip_runtime.h>
#include <stdint.h>

typedef __attribute__((ext_vector_type(2))) float v2f;
typedef __attribute__((ext_vector_type(8))) float v8f;

#define NN      4096
#define BT      128    // C block tile (128x128)
#define KT      32     // K slab staged in LDS
#define APITCH  36     // A LDS pitch (floats): 32 k + pad, 16B-aligned rows
#define BPITCH  36     // Bt LDS pitch (floats): transposed B, k contiguous
#define ASZ     (BT * APITCH)   // floats: A slab 128 rows x pitch 36
#define BSZ     (BT * BPITCH)   // floats: Bt slab 128 n-rows x pitch 36
#define BUFSZ   (ASZ + BSZ)     // floats per stage buffer

__device__ __forceinline__ uint32_t lds_off(const void* p) {
    // LDS aperture: low 32 bits of the flat address are the wave LDS byte offset.
    return (uint32_t)(uintptr_t)p;
}

__device__ __forceinline__ void async_copy_b128(uint32_t lds_byte_off, const float* gptr) {
    asm volatile("global_load_async_to_lds_b128 %0, %1, off"
                 :: "v"(lds_byte_off), "v"((uint64_t)(uintptr_t)gptr)
                 : "memory");
}

__device__ __forceinline__ void wait_async0() {
    asm volatile("s_wait_asynccnt 0x0" ::: "memory");
}

__device__ __forceinline__ void stage_slab(const float* __restrict__ A,
                                           const float* __restrict__ B,
                                           int rowBase, int colBase, int kb,
                                           float* As, float* Bt, int tid)
{
    const int tr = tid >> 3;            // A: row 0..31 within pass
    const int tc = (tid & 7) << 2;      // A: k offset 0..28 step 4
    const int bk = tid & 31;            // B: k row within slab
    const int bn = (tid >> 5) << 4;     // B: n offset 0..112 step 16

    // ---- A slab (128 x 32), row-major: mask k>=m only near A's diagonal ----
    if (kb >= rowBase + BT) {           // interior: async DMA straight to LDS
        #pragma unroll
        for (int p = 0; p < 4; ++p) {
            int r = p * 32 + tr;
            async_copy_b128(lds_off(As + r * APITCH + tc),
                            A + (size_t)(rowBase + r) * NN + kb + tc);
        }
    } else {                            // boundary: masked VGPR path
        #pragma unroll
        for (int p = 0; p < 4; ++p) {
            int r  = p * 32 + tr;
            int mg = rowBase + r;
            int kg = kb + tc;
            float4 v = *(const float4*)(A + (size_t)mg * NN + kg);
            v.x = (kg + 0 >= mg) ? v.x : 0.f;
            v.y = (kg + 1 >= mg) ? v.y : 0.f;
            v.z = (kg + 2 >= mg) ? v.z : 0.f;
            v.w = (kg + 3 >= mg) ? v.w : 0.f;
            *(float4*)(As + r * APITCH + tc) = v;
        }
    }

    // ---- B slab transposed into Bt[n][k] so frags are contiguous b64 loads.
    //      (Transpose requires the VGPR hop; mask n>=k only near B's diagonal.)
    const int kg = kb + bk;
    if (kb + KT <= colBase) {           // interior: no masking
        #pragma unroll
        for (int p = 0; p < 4; ++p) {
            int n0 = bn + p * 4;
            float4 v = *(const float4*)(B + (size_t)kg * NN + colBase + n0);
            Bt[(n0 + 0) * BPITCH + bk] = v.x;
            Bt[(n0 + 1) * BPITCH + bk] = v.y;
            Bt[(n0 + 2) * BPITCH + bk] = v.z;
            Bt[(n0 + 3) * BPITCH + bk] = v.w;
        }
    } else {                            // boundary: masked
        #pragma unroll
        for (int p = 0; p < 4; ++p) {
            int n0 = bn + p * 4;
            int ng = colBase + n0;
            float4 v = *(const float4*)(B + (size_t)kg * NN + ng);
            Bt[(n0 + 0) * BPITCH + bk] = (ng + 0 >= kg) ? v.x : 0.f;
            Bt[(n0 + 1) * BPITCH + bk] = (ng + 1 >= kg) ? v.y : 0.f;
            Bt[(n0 + 2) * BPITCH + bk] = (ng + 2 >= kg) ? v.z : 0.f;
            Bt[(n0 + 3) * BPITCH + bk] = (ng + 3 >= kg) ? v.w : 0.f;
        }
    }
}

__device__ __forceinline__ void compute_slab(const float* As, const float* Bt,
                                             int waveM, int waveN, int lh, int ll,
                                             v8f acc[2][4])
{
    #pragma unroll
    for (int kk = 0; kk < KT; kk += 4) {
        const int kq = kk + 2 * lh;     // lanes 0-15: K pair {k,k+1}; 16-31: {k+2,k+3}
        v2f af[2], bf[4];
        #pragma unroll
        for (int mi = 0; mi < 2; ++mi) {
            int m = waveM * 32 + mi * 16 + ll;
            af[mi] = *(const v2f*)(As + m * APITCH + kq);
        }
        #pragma unroll
        for (int ni = 0; ni < 4; ++ni) {
            int n = waveN * 64 + ni * 16 + ll;
            bf[ni] = *(const v2f*)(Bt + n * BPITCH + kq);
        }
        #pragma unroll
        for (int mi = 0; mi < 2; ++mi)
            #pragma unroll
            for (int ni = 0; ni < 4; ++ni)
                acc[mi][ni] = __builtin_amdgcn_wmma_f32_16x16x4_f32(
                    false, af[mi], false, bf[ni],
                    (short)0, acc[mi][ni], false, false);
    }
}

__global__ __launch_bounds__(256) void triu_gemm_wmma_f32(
    const float* __restrict__ A, const float* __restrict__ B,
    float* __restrict__ C)
{
    const int bRow    = blockIdx.y;
    const int bCol    = blockIdx.x;
    const int rowBase = bRow * BT;
    const int colBase = bCol * BT;
    const int tid     = threadIdx.x;

    // Strictly-lower blocks: output is exactly zero (must write: d_out is poisoned).
    if (bRow > bCol) {
        for (int p = 0; p < 16; ++p) {
            int c   = p * 256 + tid;
            int r   = c >> 5;
            int col = (c & 31) << 2;
            *(float4*)(C + (size_t)(rowBase + r) * NN + colBase + col) =
                make_float4(0.f, 0.f, 0.f, 0.f);
        }
        return;
    }

    __shared__ float lds[2 * BUFSZ];

    const int wave  = tid >> 5;
    const int lane  = tid & 31;
    const int lh    = lane >> 4;
    const int ll    = lane & 15;
    const int waveM = wave & 3;     // rows waveM*32 .. +31
    const int waveN = wave >> 2;    // cols waveN*64 .. +63

    v8f acc[2][4];
    #pragma unroll
    for (int mi = 0; mi < 2; ++mi)
        #pragma unroll
        for (int ni = 0; ni < 4; ++ni)
            acc[mi][ni] = (v8f){};

    const int kEnd = colBase + BT;

    // Prologue: stage first slab into buffer 0. (LDS pointers computed
    // dynamically — never placed in an initialized array.)
    stage_slab(A, B, rowBase, colBase, rowBase, lds, lds + ASZ, tid);
    wait_async0();
    __syncthreads();

    int cur = 0;
    for (int kb = rowBase; kb < kEnd; kb += KT) {
        const int nxt = cur ^ 1;
        float* curBase = lds + cur * BUFSZ;
        float* nxtBase = lds + nxt * BUFSZ;

        if (kb + KT < kEnd)   // prefetch next slab into the other buffer
            stage_slab(A, B, rowBase, colBase, kb + KT,
                       nxtBase, nxtBase + ASZ, tid);

        compute_slab(curBase, curBase + ASZ, waveM, waveN, lh, ll, acc);

        wait_async0();        // my async LDS writes for 'nxt' complete
        __syncthreads();      // everyone's writes visible; reads of 'cur' done
        cur = nxt;
    }

    // Epilogue: D layout => VGPR j: row = tile + j + 8*lh, col = tile + ll.
    for (int mi = 0; mi < 2; ++mi) {
        int tileR = rowBase + waveM * 32 + mi * 16;
        for (int ni = 0; ni < 4; ++ni) {
            int tileC = colBase + waveN * 64 + ni * 16;
            int col   = tileC + ll;
            #pragma unroll
            for (int j = 0; j < 8; ++j) {
                int row = tileR + j + 8 * lh;
                float v = (col >= row) ? acc[mi][ni][j] : 0.f;
                C[(size_t)row * NN + col] = v;
            }
        }
    }
}

extern "C" void kernel_launch(void* const* d_in, const int* in_sizes, int n_in,
                              void* d_out, int out_size, void* d_ws, size_t ws_size,
                              hipStream_t stream) {
    (void)in_sizes; (void)n_in; (void)out_size; (void)d_ws; (void)ws_size;
    const float* A = (const float*)d_in[0];
    const float* B = (const float*)d_in[1];
    float* C = (float*)d_out;
    dim3 grid(NN / BT, NN / BT);   // x = column block, y = row block
    triu_gemm_wmma_f32<<<grid, 256, 0, stream>>>(A, B, C);
}